// Cam_59785944760667
// MI455X (gfx1250) — compile-verified
//
#include <hip/hip_runtime.h>
#include <hip/hip_bf16.h>

// Problem constants (match reference)
#define NN 100000
#define DD 128
#define HH 64
#define KK 8
#define LL 3
#define CC 16
#define EE 1600000
#define THETA 0.1f

typedef __attribute__((ext_vector_type(2))) float v2f;
typedef __attribute__((ext_vector_type(8))) float v8f;

// D = A(16x4,f32) * B(4x16,f32) + C(16x16,f32)  -> v_wmma_f32_16x16x4_f32
__device__ __forceinline__ v8f wmma_f32(v2f a, v2f b, v8f c) {
  return __builtin_amdgcn_wmma_f32_16x16x4_f32(
      /*neg_a=*/false, a, /*neg_b=*/false, b,
      /*c_mod=*/(short)0, c, /*reuse_a=*/false, /*reuse_b=*/false);
}

__global__ void zero_kernel(float* __restrict__ p, int n) {
  int i = blockIdx.x * blockDim.x + threadIdx.x;
  if (i < n) p[i] = 0.0f;
}

__global__ void degree_kernel(const int* __restrict__ col, float* __restrict__ deg) {
  int e = blockIdx.x * blockDim.x + threadIdx.x;
  if (e < EE) atomicAdd(&deg[col[e]], 1.0f);
}

__global__ void dn_kernel(const float* __restrict__ deg, float* __restrict__ dn) {
  int i = blockIdx.x * blockDim.x + threadIdx.x;
  if (i < NN) {
    float d = deg[i];
    dn[i] = (d > 0.0f) ? rsqrtf(d) : 0.0f;
  }
}

// h = relu(x @ fc0_w + fc0_b) : [N,128] x [128,64]
// One wave -> 16-node x 64-col strip. 8 waves / block -> 128 nodes per block.
__global__ __launch_bounds__(256) void fc0_kernel(const float* __restrict__ x,
                                                  const float* __restrict__ w,
                                                  const float* __restrict__ b,
                                                  float* __restrict__ h) {
  const int wave = threadIdx.x >> 5;
  const int lane = threadIdx.x & 31;
  const int half = lane >> 4;   // selects K pair {0,1} vs {2,3} within a chunk
  const int lid  = lane & 15;   // row (A) / col (B,C)
  const int node0 = blockIdx.x * 128 + wave * 16;
  if (node0 >= NN) return;      // N % 16 == 0, so live waves are full tiles

  v8f acc0 = {}, acc1 = {}, acc2 = {}, acc3 = {};
  for (int c = 0; c < DD / 4; ++c) {
    const int k0 = 4 * c + 2 * half;
    const float* ap = x + (node0 + lid) * DD + k0;
    v2f a; a.x = ap[0]; a.y = ap[1];
    const float* bp = w + k0 * HH;           // rows k0, k0+1 of [128][64]
    v2f b0; b0.x = bp[lid];      b0.y = bp[HH + lid];
    v2f b1; b1.x = bp[16 + lid]; b1.y = bp[HH + 16 + lid];
    v2f b2; b2.x = bp[32 + lid]; b2.y = bp[HH + 32 + lid];
    v2f b3; b3.x = bp[48 + lid]; b3.y = bp[HH + 48 + lid];
    acc0 = wmma_f32(a, b0, acc0);
    acc1 = wmma_f32(a, b1, acc1);
    acc2 = wmma_f32(a, b2, acc2);
    acc3 = wmma_f32(a, b3, acc3);
  }
  v8f accs[4] = {acc0, acc1, acc2, acc3};
  for (int t = 0; t < 4; ++t) {
    const float bias = b[16 * t + lid];
    for (int r = 0; r < 8; ++r) {
      float v = accs[t][r] + bias;
      h[(node0 + r + 8 * half) * HH + 16 * t + lid] = v > 0.0f ? v : 0.0f;
    }
  }
}

// Per-edge scatter: agg[col] += dn[row]*dn[col] * h[row]
// One thread per (edge, 16-byte feature chunk): 16 chunks cover H=64 floats.
__global__ void scatter_kernel(const int* __restrict__ row, const int* __restrict__ col,
                               const float* __restrict__ dn, const float* __restrict__ h,
                               float* __restrict__ agg) {
  unsigned idx = blockIdx.x * blockDim.x + threadIdx.x;   // EE*16 = 25.6M items
  unsigned e = idx >> 4;
  int seg = idx & 15;
  if (e >= EE) return;
  int r = row[e], c = col[e];
  float v = dn[r] * dn[c];
  const float4 hv = *(const float4*)(h + r * HH + seg * 4);
  float* ap = agg + c * HH + seg * 4;
  atomicAdd(ap + 0, v * hv.x);
  atomicAdd(ap + 1, v * hv.y);
  atomicAdd(ap + 2, v * hv.z);
  atomicAdd(ap + 3, v * hv.w);
}

// e = pi * (pi > theta), pi = softmax(h @ env_w[layer][:H] + env_b[layer])
// (h_env is identically zero, so rows H..2H-1 of env_w contribute nothing)
__global__ void gate_kernel(const float* __restrict__ h, const float* __restrict__ env_w,
                            const float* __restrict__ env_b, float* __restrict__ e_out,
                            int layer) {
  int n = blockIdx.x * blockDim.x + threadIdx.x;
  if (n >= NN) return;
  const float* W = env_w + layer * 2 * HH * KK;   // [2H][K]
  const float* B = env_b + layer * KK;
  float logit[KK];
  #pragma unroll
  for (int k = 0; k < KK; ++k) logit[k] = B[k];
  for (int f = 0; f < HH; ++f) {
    float hv = h[n * HH + f];
    const float* wr = W + f * KK;
    #pragma unroll
    for (int k = 0; k < KK; ++k) logit[k] += hv * wr[k];
  }
  float m = logit[0];
  #pragma unroll
  for (int k = 1; k < KK; ++k) m = fmaxf(m, logit[k]);
  float s = 0.0f;
  float ex[KK];
  #pragma unroll
  for (int k = 0; k < KK; ++k) { ex[k] = __expf(logit[k] - m); s += ex[k]; }
  float inv = 1.0f / s;
  #pragma unroll
  for (int k = 0; k < KK; ++k) {
    float pi = ex[k] * inv;
    e_out[n * KK + k] = (pi > THETA) ? pi : 0.0f;
  }
}

// h_out = relu( sum_k e[n,k] * ([agg||h] @ conv_w[layer][k]) + h )
// Implemented by scaling A-fragment rows by e[n,k] and accumulating over k.
__global__ __launch_bounds__(256) void conv_kernel(const float* __restrict__ agg,
                                                   const float* __restrict__ h,
                                                   const float* __restrict__ e_gate,
                                                   const float* __restrict__ conv_w,
                                                   float* __restrict__ h_out,
                                                   int layer) {
  const int wave = threadIdx.x >> 5;
  const int lane = threadIdx.x & 31;
  const int half = lane >> 4;
  const int lid  = lane & 15;
  const int node0 = blockIdx.x * 128 + wave * 16;
  if (node0 >= NN) return;

  const float* W = conv_w + layer * (KK * 2 * HH * HH);   // [K][128][64]
  v8f acc0 = {}, acc1 = {}, acc2 = {}, acc3 = {};

  #pragma unroll 1
  for (int k = 0; k < KK; ++k) {
    const float ek = e_gate[(node0 + lid) * KK + k];      // row gate for A row lid
    if (!__any(ek != 0.0f)) continue;                     // wave-uniform skip
    const float* Wk = W + k * (2 * HH * HH);
    #pragma unroll 1
    for (int c = 0; c < (2 * HH) / 4; ++c) {              // 32 K-chunks over hi
      const int kk = 4 * c + 2 * half;                    // row pair in [0,128)
      const float* src = (kk < HH) ? agg : h;             // chunk lies fully in one half
      const int fk = (kk < HH) ? kk : kk - HH;
      const float* ap = src + (node0 + lid) * HH + fk;
      v2f a; a.x = ek * ap[0]; a.y = ek * ap[1];
      const float* bp = Wk + kk * HH;                     // rows kk, kk+1 of [128][64]
      v2f b0; b0.x = bp[lid];      b0.y = bp[HH + lid];
      v2f b1; b1.x = bp[16 + lid]; b1.y = bp[HH + 16 + lid];
      v2f b2; b2.x = bp[32 + lid]; b2.y = bp[HH + 32 + lid];
      v2f b3; b3.x = bp[48 + lid]; b3.y = bp[HH + 48 + lid];
      acc0 = wmma_f32(a, b0, acc0);
      acc1 = wmma_f32(a, b1, acc1);
      acc2 = wmma_f32(a, b2, acc2);
      acc3 = wmma_f32(a, b3, acc3);
    }
  }
  v8f accs[4] = {acc0, acc1, acc2, acc3};
  for (int t = 0; t < 4; ++t) {
    for (int r = 0; r < 8; ++r) {
      const int idx = (node0 + r + 8 * half) * HH + 16 * t + lid;
      float v = accs[t][r] + h[idx];                      // residual
      h_out[idx] = v > 0.0f ? v : 0.0f;                   // relu
    }
  }
}

// out = h @ fc1_w + fc1_b : [N,64] x [64,16]
__global__ __launch_bounds__(256) void fc1_kernel(const float* __restrict__ h,
                                                  const float* __restrict__ w,
                                                  const float* __restrict__ b,
                                                  float* __restrict__ out) {
  const int wave = threadIdx.x >> 5;
  const int lane = threadIdx.x & 31;
  const int half = lane >> 4;
  const int lid  = lane & 15;
  const int node0 = blockIdx.x * 128 + wave * 16;
  if (node0 >= NN) return;

  v8f acc = {};
  for (int c = 0; c < HH / 4; ++c) {
    const int k0 = 4 * c + 2 * half;
    const float* ap = h + (node0 + lid) * HH + k0;
    v2f a; a.x = ap[0]; a.y = ap[1];
    const float* bp = w + k0 * CC;                        // rows k0, k0+1 of [64][16]
    v2f bb; bb.x = bp[lid]; bb.y = bp[CC + lid];
    acc = wmma_f32(a, bb, acc);
  }
  const float bias = b[lid];
  for (int r = 0; r < 8; ++r)
    out[(node0 + r + 8 * half) * CC + lid] = acc[r] + bias;
}

extern "C" void kernel_launch(void* const* d_in, const int* in_sizes, int n_in,
                              void* d_out, int out_size, void* d_ws, size_t ws_size,
                              hipStream_t stream) {
  const float* x      = (const float*)d_in[0];
  const int*   eidx   = (const int*)d_in[1];
  const float* fc0_w  = (const float*)d_in[2];
  const float* fc0_b  = (const float*)d_in[3];
  const float* fc1_w  = (const float*)d_in[4];
  const float* fc1_b  = (const float*)d_in[5];
  const float* env_w  = (const float*)d_in[6];
  const float* env_b  = (const float*)d_in[7];
  const float* conv_w = (const float*)d_in[8];
  float* out = (float*)d_out;

  const int* row = eidx;        // edge_index[0]
  const int* col = eidx + EE;   // edge_index[1]

  // Workspace layout (floats): deg[N] | dn[N] | hA[N*H] | hB[N*H] | agg[N*H] | e[N*K]
  float* ws  = (float*)d_ws;
  float* deg = ws;
  float* dn  = deg + NN;
  float* hA  = dn + NN;
  float* hB  = hA + (size_t)NN * HH;
  float* agg = hB + (size_t)NN * HH;
  float* eg  = agg + (size_t)NN * HH;

  const int T = 256;
  const int gemm_blocks = (NN + 127) / 128;   // 8 waves x 16 nodes per block

  // Degree + symmetric norm (graph is fixed across layers)
  zero_kernel<<<(NN + T - 1) / T, T, 0, stream>>>(deg, NN);
  degree_kernel<<<(EE + T - 1) / T, T, 0, stream>>>(col, deg);
  dn_kernel<<<(NN + T - 1) / T, T, 0, stream>>>(deg, dn);

  // h = relu(x @ fc0_w + fc0_b)
  fc0_kernel<<<gemm_blocks, T, 0, stream>>>(x, fc0_w, fc0_b, hA);

  float* cur = hA;
  float* nxt = hB;
  for (int layer = 0; layer < LL; ++layer) {
    zero_kernel<<<(NN * HH + T - 1) / T, T, 0, stream>>>(agg, NN * HH);
    {
      unsigned items = (unsigned)EE * 16u;
      scatter_kernel<<<(items + T - 1) / T, T, 0, stream>>>(row, col, dn, cur, agg);
    }
    gate_kernel<<<(NN + T - 1) / T, T, 0, stream>>>(cur, env_w, env_b, eg, layer);
    conv_kernel<<<gemm_blocks, T, 0, stream>>>(agg, cur, eg, conv_w, nxt, layer);
    float* tmp = cur; cur = nxt; nxt = tmp;
  }

  // out = h @ fc1_w + fc1_b
  fc1_kernel<<<gemm_blocks, T, 0, stream>>>(cur, fc1_w, fc1_b, out);
}